// GraphSAGERecommender_53360673685665
// MI455X (gfx1250) — compile-verified
//
#include <hip/hip_runtime.h>
#include <math.h>

// CDNA5 wave32 WMMA fragment types (f32 16x16x4: A=16x4 -> 2 VGPR, B=4x16 -> 2 VGPR, C/D=16x16 -> 8 VGPR)
typedef float v2f __attribute__((ext_vector_type(2)));
typedef float v8f __attribute__((ext_vector_type(8)));

// Only the device pass has amdgcn builtins; the host pass must not evaluate this.
#if defined(__HIP_DEVICE_COMPILE__) && defined(__AMDGCN__)
#if !__has_builtin(__builtin_amdgcn_wmma_f32_16x16x4_f32)
#error "missing __builtin_amdgcn_wmma_f32_16x16x4_f32 on this toolchain"
#endif
#endif

// ---------------------------------------------------------------------------
// Edge scatter: agg[dst] += feat[src] (and optionally deg[dst] += 1).
// chunks = F/4 lanes per edge, each lane moves one float4 of the row.
// x (25.6MB) / h (51.2MB) are L2-resident on MI455X (192MB L2), so these are
// L2 gathers + L2 f32 atomics; this pass dominates total runtime.
// ---------------------------------------------------------------------------
__global__ void sage_edge_scatter(const long long* __restrict__ ei,
                                  const float* __restrict__ feat,
                                  float* __restrict__ agg,
                                  float* __restrict__ deg,
                                  int E, int F, int chunkShift) {
  long long tid = (long long)blockIdx.x * blockDim.x + threadIdx.x;
  long long e = tid >> chunkShift;
  int c = (int)(tid & ((1 << chunkShift) - 1));
  if (e >= E) return;
  long long s = ei[e];                 // edge_index row 0 = src
  long long d = ei[(long long)E + e];  // edge_index row 1 = dst
  const float4 v = reinterpret_cast<const float4*>(feat + s * (long long)F)[c];
  float* o = agg + d * (long long)F + (long long)c * 4;
  atomicAdd(o + 0, v.x);
  atomicAdd(o + 1, v.y);
  atomicAdd(o + 2, v.z);
  atomicAdd(o + 3, v.w);
  if (deg != nullptr && c == 0) atomicAdd(deg + d, 1.0f);
}

// ---------------------------------------------------------------------------
// Fused SAGE layer GEMM via V_WMMA_F32_16X16X4_F32:
//   out[m][n] = act( (agg[m]*inv_deg[m]) @ Wl^T + xin[m] @ Wr^T + bias[n] )
// K1/NOUT/RELU are compile-time so both K-halves are branch-free and fully
// unrolled; two independent accumulator chains keep the WMMA pipe busy.
// One wave computes one 16x16 tile, blockDim must be 256 (8 waves/block).
// Fragment layouts (ISA 7.12.2, 32-bit, wave32):
//   A 16x4 : lane&15 = M row, (lane>>4)*2 = K base; v2f = {K, K+1}
//   B 4x16 : lane&15 = N col, (lane>>4)*2 = K base; v2f = {K, K+1}
//   D 16x16: VGPR i -> M = (lane>>4)*8 + i, N = lane&15
// ---------------------------------------------------------------------------
template <int K1, int NOUT, bool RELU>
__global__ void sage_wmma_gemm(const float* __restrict__ agg,
                               const float* __restrict__ xin,
                               const float* __restrict__ deg,
                               const float* __restrict__ Wl,
                               const float* __restrict__ Wr,
                               const float* __restrict__ bias,
                               float* __restrict__ out,
                               int M) {
  const int lane = threadIdx.x & 31;
  const int wave = threadIdx.x >> 5;
  constexpr int nTilesN = NOUT >> 4;
  const int mTiles = (M + 15) >> 4;
  const int tile = blockIdx.x * 8 + wave;
  if (tile >= mTiles * nTilesN) return;   // whole wave exits together
  const int mTile = tile / nTilesN;
  const int nTile = tile - mTile * nTilesN;
  const int mBase = mTile << 4;
  const int nBase = nTile << 4;

  const int r = lane & 15;
  const int koff = (lane >> 4) << 1;      // 0 or 2

  int m = mBase + r;
  if (m >= M) m = M - 1;                  // clamp loads, keep EXEC all-ones for WMMA
  const int n = nBase + r;

  const float inv = 1.0f / fmaxf(deg[m], 1.0f);

  const float* aggRow = agg + (long long)m * K1;
  const float* xRow   = xin + (long long)m * K1;
  const float* wlRow  = Wl  + (long long)n * K1;
  const float* wrRow  = Wr  + (long long)n * K1;

  v8f acc0 = {};   // (agg * inv_deg) @ Wl^T
  v8f acc1 = {};   // xin @ Wr^T
#pragma unroll
  for (int k = 0; k < K1; k += 4) {
    const int kk = k + koff;              // kk even -> 8B-aligned float2 loads
    const float2 av = *reinterpret_cast<const float2*>(aggRow + kk);
    const float2 bv = *reinterpret_cast<const float2*>(wlRow + kk);
    v2f a, b;
    a.x = av.x * inv; a.y = av.y * inv;
    b.x = bv.x;       b.y = bv.y;
    acc0 = __builtin_amdgcn_wmma_f32_16x16x4_f32(false, a, false, b,
                                                 (short)0, acc0, false, false);
  }
#pragma unroll
  for (int k = 0; k < K1; k += 4) {
    const int kk = k + koff;
    const float2 av = *reinterpret_cast<const float2*>(xRow + kk);
    const float2 bv = *reinterpret_cast<const float2*>(wrRow + kk);
    v2f a, b;
    a.x = av.x; a.y = av.y;
    b.x = bv.x; b.y = bv.y;
    acc1 = __builtin_amdgcn_wmma_f32_16x16x4_f32(false, a, false, b,
                                                 (short)0, acc1, false, false);
  }

  const float bs = bias[n];
  const int rowOff = (lane >> 4) << 3;    // 0 or 8
#pragma unroll
  for (int i = 0; i < 8; ++i) {
    const int mr = mBase + rowOff + i;
    if (mr < M) {
      float v = acc0[i] + acc1[i] + bs;
      if (RELU) v = fmaxf(v, 0.0f);
      out[(long long)mr * NOUT + n] = v;
    }
  }
}

// ---------------------------------------------------------------------------
// Link predictor factorization: s1[m] = dot(h2[m], Wlp[0:64]),
//                               s2[m] = dot(h2[m], Wlp[64:128])
// so each pair only needs two scalar loads instead of a 512B gather.
// ---------------------------------------------------------------------------
__global__ void sage_node_scores(const float* __restrict__ h2,
                                 const float* __restrict__ Wlp,
                                 float* __restrict__ s12, int N) {
  __shared__ float w[128];
  if (threadIdx.x < 128) w[threadIdx.x] = Wlp[threadIdx.x];
  __syncthreads();
  const int m = blockIdx.x * blockDim.x + threadIdx.x;
  if (m >= N) return;
  const float4* row = reinterpret_cast<const float4*>(h2 + (long long)m * 64);
  float s1 = 0.0f, s2 = 0.0f;
#pragma unroll
  for (int j = 0; j < 16; ++j) {
    const float4 v = row[j];
    s1 += v.x * w[4 * j + 0] + v.y * w[4 * j + 1] + v.z * w[4 * j + 2] + v.w * w[4 * j + 3];
    s2 += v.x * w[64 + 4 * j + 0] + v.y * w[64 + 4 * j + 1] +
          v.z * w[64 + 4 * j + 2] + v.w * w[64 + 4 * j + 3];
  }
  s12[2 * m + 0] = s1;
  s12[2 * m + 1] = s2;
}

__global__ void sage_pair_predict(const long long* __restrict__ pairs,
                                  const float* __restrict__ s12,
                                  const float* __restrict__ blp,
                                  float* __restrict__ out, int P) {
  const int p = blockIdx.x * blockDim.x + threadIdx.x;
  if (p >= P) return;
  const long long s = pairs[2 * (long long)p + 0];
  const long long d = pairs[2 * (long long)p + 1];
  const float z = s12[2 * s + 0] + s12[2 * d + 1] + blp[0];
  out[p] = 1.0f / (1.0f + expf(-z));
}

// ---------------------------------------------------------------------------
// Inputs (setup_inputs order):
// 0:x(100000x64 f32) 1:edge_index(2x1.6M i64) 2:pairs(200000x2 i64)
// 3:W1l(128x64) 4:W1r(128x64) 5:b1(128) 6:W2l(64x128) 7:W2r(64x128) 8:b2(64)
// 9:Wlp(1x128) 10:blp(1)
// ---------------------------------------------------------------------------
extern "C" void kernel_launch(void* const* d_in, const int* in_sizes, int n_in,
                              void* d_out, int out_size, void* d_ws, size_t ws_size,
                              hipStream_t stream) {
  const float*     x     = (const float*)d_in[0];
  const long long* ei    = (const long long*)d_in[1];
  const long long* pairs = (const long long*)d_in[2];
  const float*     W1l   = (const float*)d_in[3];
  const float*     W1r   = (const float*)d_in[4];
  const float*     b1    = (const float*)d_in[5];
  const float*     W2l   = (const float*)d_in[6];
  const float*     W2r   = (const float*)d_in[7];
  const float*     b2    = (const float*)d_in[8];
  const float*     Wlp   = (const float*)d_in[9];
  const float*     blp   = (const float*)d_in[10];
  float* out = (float*)d_out;

  const int N = in_sizes[0] / 64;   // 100000 nodes
  const int E = in_sizes[1] / 2;    // 1.6M edges
  const int P = in_sizes[2] / 2;    // 200k pairs

  // Workspace layout (floats). agg1 is dead after GEMM1 -> reuse as h2;
  // agg2 is dead after GEMM2 -> reuse as per-node score pairs.
  float* ws   = (float*)d_ws;
  float* deg  = ws;                           // N
  float* agg1 = deg  + N;                     // N*64
  float* h    = agg1 + (size_t)N * 64;        // N*128
  float* agg2 = h    + (size_t)N * 128;       // N*128
  float* h2   = agg1;                         // N*64  (alias)
  float* s12  = agg2;                         // N*2   (alias)
  (void)ws_size; (void)n_in; (void)out_size;

  // Zero the accumulators every call (harness does not re-poison between replays).
  (void)hipMemsetAsync(deg,  0, sizeof(float) * (size_t)N * 65, stream);   // deg + agg1 contiguous
  (void)hipMemsetAsync(agg2, 0, sizeof(float) * (size_t)N * 128, stream);

  // ---- Layer 1: mean-aggregate + fused dual GEMM + ReLU ----
  {
    const long long work = (long long)E * 16;             // 16 lanes/edge (64 feats)
    const int blocks = (int)((work + 255) / 256);
    sage_edge_scatter<<<blocks, 256, 0, stream>>>(ei, x, agg1, deg, E, 64, 4);
  }
  {
    const int tiles = ((N + 15) / 16) * (128 / 16);
    sage_wmma_gemm<64, 128, true><<<(tiles + 7) / 8, 256, 0, stream>>>(
        agg1, x, deg, W1l, W1r, b1, h, N);
  }

  // ---- Layer 2: mean-aggregate + fused dual GEMM ----
  {
    const long long work = (long long)E * 32;             // 32 lanes/edge (128 feats)
    const int blocks = (int)((work + 255) / 256);
    sage_edge_scatter<<<blocks, 256, 0, stream>>>(ei, h, agg2, nullptr, E, 128, 5);
  }
  {
    const int tiles = ((N + 15) / 16) * (64 / 16);
    sage_wmma_gemm<128, 64, false><<<(tiles + 7) / 8, 256, 0, stream>>>(
        agg2, h, deg, W2l, W2r, b2, h2, N);
  }

  // ---- Link predictor ----
  sage_node_scores<<<(N + 255) / 256, 256, 0, stream>>>(h2, Wlp, s12, N);
  sage_pair_predict<<<(P + 255) / 256, 256, 0, stream>>>(pairs, s12, blp, out, P);
}